// UHGSAGEConv_59322088292912
// MI455X (gfx1250) — compile-verified
//
#include <hip/hip_runtime.h>

typedef __attribute__((ext_vector_type(2))) float v2f;
typedef __attribute__((ext_vector_type(8))) float v8f;

#define IN_F  128
#define OUT_F 128
#define KDIM  256   // 2*IN_F

// ---------------------------------------------------------------- zero agg+cnt
__global__ void zero_ws_kernel(float* __restrict__ p, long long n) {
    long long i = (long long)blockIdx.x * blockDim.x + threadIdx.x;
    long long stride = (long long)gridDim.x * blockDim.x;
    for (; i < n; i += stride) p[i] = 0.0f;
}

// --------------------------------------------------- edge scatter-add (mean prep)
// one edge per wave32: lane handles 4 contiguous floats (float4 load, 4 f32 atomics)
__global__ void scatter_add_kernel(const float* __restrict__ x,
                                   const int* __restrict__ src,
                                   const int* __restrict__ tgt,
                                   float* __restrict__ agg,
                                   float* __restrict__ cnt,
                                   int n_edges) {
    long long t = (long long)blockIdx.x * blockDim.x + threadIdx.x;
    int e    = (int)(t >> 5);
    int lane = (int)(t & 31);
    if (e >= n_edges) return;
    int s = src[e];
    int d = tgt[e];
    const float4 v = *(const float4*)(x + (long long)s * IN_F + lane * 4);
    float* dst = agg + (long long)d * IN_F + lane * 4;
    atomicAdd(dst + 0, v.x);
    atomicAdd(dst + 1, v.y);
    atomicAdd(dst + 2, v.z);
    atomicAdd(dst + 3, v.w);
    if (lane == 0) atomicAdd(cnt + d, 1.0f);
}

// ---------------------------------------------------------------- agg /= max(cnt,1)
__global__ void scale_agg_kernel(float* __restrict__ agg,
                                 const float* __restrict__ cnt,
                                 long long total) {
    long long i = (long long)blockIdx.x * blockDim.x + threadIdx.x;
    if (i >= total) return;
    int m = (int)(i >> 7);                 // /128
    float c = fmaxf(cnt[m], 1.0f);
    agg[i] = agg[i] / c;
}

// ---------------------------------------------------------------- WMMA f32 GEMM
// out_tmp[m][n] = relu( sum_k h[m][k]*W[n][k] + b[n] ),  h = [x | agg]
// block = 256 threads = 8 waves; block -> one 16-row M tile; wave w -> N tile w.
// K loop statically split at the x|agg boundary (kb is a multiple of 4, so the
// boundary is never straddled) -> no per-lane address selects in the hot loop.
__global__ void __launch_bounds__(256)
gemm_wmma_f32_kernel(const float* __restrict__ x,
                     const float* __restrict__ agg,
                     const float* __restrict__ W,
                     const float* __restrict__ bias,
                     float* __restrict__ out_tmp,
                     int n_rows) {
    const int wave = threadIdx.x >> 5;      // 0..7 -> N tile
    const int lane = threadIdx.x & 31;
    const int lo16 = lane & 15;
    const int hi   = lane >> 4;             // 0 or 1: K sub-pair
    const int tile_m = blockIdx.x;

    int row = tile_m * 16 + lo16;
    int rc  = row < n_rows ? row : (n_rows - 1);   // clamp; OOB rows not stored
    const int col = wave * 16 + lo16;              // output feature (B fragment N)

    const float* xrow = x   + (long long)rc  * IN_F + 2 * hi;
    const float* arow = agg + (long long)rc  * IN_F + 2 * hi;
    const float* wrow = W   + (long long)col * KDIM + 2 * hi;

    v8f c = {};
    // first half of K: h[:, 0:128] = x
#pragma unroll 8
    for (int kb = 0; kb < IN_F; kb += 4) {
        v2f a = *(const v2f*)(xrow + kb);
        v2f b = *(const v2f*)(wrow + kb);
        c = __builtin_amdgcn_wmma_f32_16x16x4_f32(false, a, false, b,
                                                  (short)0, c, false, false);
    }
    // second half of K: h[:, 128:256] = agg (mean-aggregated)
#pragma unroll 8
    for (int kb = 0; kb < IN_F; kb += 4) {
        v2f a = *(const v2f*)(arow + kb);
        v2f b = *(const v2f*)(wrow + IN_F + kb);
        c = __builtin_amdgcn_wmma_f32_16x16x4_f32(false, a, false, b,
                                                  (short)0, c, false, false);
    }

    // bias + relu epilogue. C/D layout: VGPR i, lane l -> [M=i+8*(l>>4)][N=l&15]
    const float bv = bias[col];
    float* outp = out_tmp + (long long)(tile_m * 16 + 8 * hi) * OUT_F + col;
    if (tile_m * 16 + 16 <= n_rows) {          // uniform: full tile in range
#pragma unroll
        for (int i = 0; i < 8; ++i)
            outp[(long long)i * OUT_F] = fmaxf(c[i] + bv, 0.0f);
    } else {                                   // last (ragged) tile only
#pragma unroll
        for (int i = 0; i < 8; ++i) {
            int m = tile_m * 16 + i + 8 * hi;
            if (m < n_rows)
                outp[(long long)i * OUT_F] = fmaxf(c[i] + bv, 0.0f);
        }
    }
}

// ------------------------------------------------ row normalize + append hom=1
// one wave per row; lane holds 4 floats; faithful two-stage UHG normalization.
__global__ void normalize_kernel(const float* __restrict__ tmp,
                                 float* __restrict__ out,
                                 int n_rows) {
    long long t = (long long)blockIdx.x * blockDim.x + threadIdx.x;
    int row  = (int)(t >> 5);
    int lane = (int)(t & 31);
    if (row >= n_rows) return;

    float4 v = *(const float4*)(tmp + (long long)row * OUT_F + lane * 4);
    float ss = v.x * v.x + v.y * v.y + v.z * v.z + v.w * v.w;
#pragma unroll
    for (int off = 16; off > 0; off >>= 1) ss += __shfl_xor(ss, off, 32);
    const float S = ss;                         // sum feats^2 ; hom == 1

    // _uhg_normalize: norm1 = sqrt(clip(S - 1, 1e-8)); feats1 = f/norm1; hom=1
    float norm1 = sqrtf(fmaxf(S - 1.0f, 1e-8f));
    bool  zero  = (S == 0.0f);                  // relu>=0 -> S==0 iff all feats 0

    // _normalize_points: zero rows -> ones; n = ||feats1||; feats /= clip(n,1e-8)
    float nn  = zero ? sqrtf(128.0f) : (sqrtf(S) / norm1);
    float inv = 1.0f / fmaxf(nn, 1e-8f);

    float4 o;
    if (zero) {
        o.x = o.y = o.z = o.w = inv;            // ones / sqrt(128)
    } else {
        float sc = inv / norm1;                 // == 1/sqrt(S) analytically
        o.x = v.x * sc; o.y = v.y * sc; o.z = v.z * sc; o.w = v.w * sc;
    }
    float* orow = out + (long long)row * (OUT_F + 1) + lane * 4;
    orow[0] = o.x; orow[1] = o.y; orow[2] = o.z; orow[3] = o.w;
    if (lane == 0) out[(long long)row * (OUT_F + 1) + OUT_F] = 1.0f;  // hom, sign=+1
}

// ------------------------------------------------------------------- launcher
extern "C" void kernel_launch(void* const* d_in, const int* in_sizes, int n_in,
                              void* d_out, int out_size, void* d_ws, size_t ws_size,
                              hipStream_t stream) {
    const float* x    = (const float*)d_in[0];
    const float* W    = (const float*)d_in[1];
    const float* bias = (const float*)d_in[2];
    const int*   ei   = (const int*)d_in[3];

    const int N = in_sizes[0] / IN_F;
    const int E = in_sizes[3] / 2;
    const int* src = ei;          // edge_index[0]
    const int* tgt = ei + E;      // edge_index[1]

    float* ws      = (float*)d_ws;
    float* agg     = ws;                                  // N*128
    float* cnt     = ws + (long long)N * IN_F;            // N
    float* out_tmp = cnt + N;                             // N*128
    float* out     = (float*)d_out;                       // N*129

    // 1) zero agg + cnt (contiguous)
    {
        long long nz = (long long)N * IN_F + N;
        zero_ws_kernel<<<2048, 256, 0, stream>>>(agg, nz);
    }
    // 2) scatter-add over edges (wave per edge)
    {
        long long threads = (long long)E * 32;
        int blocks = (int)((threads + 255) / 256);
        scatter_add_kernel<<<blocks, 256, 0, stream>>>(x, src, tgt, agg, cnt, E);
    }
    // 3) mean: agg /= max(cnt,1)
    {
        long long total = (long long)N * IN_F;
        int blocks = (int)((total + 255) / 256);
        scale_agg_kernel<<<blocks, 256, 0, stream>>>(agg, cnt, total);
    }
    // 4) WMMA fp32 GEMM + bias + relu
    {
        int tiles_m = (N + 15) / 16;
        gemm_wmma_f32_kernel<<<tiles_m, 256, 0, stream>>>(x, agg, W, bias, out_tmp, N);
    }
    // 5) row normalization + homogeneous coordinate
    {
        long long threads = (long long)N * 32;
        int blocks = (int)((threads + 255) / 256);
        normalize_kernel<<<blocks, 256, 0, stream>>>(out_tmp, out, N);
    }
}